// ComplexLayerNorm1d_62732292325506
// MI455X (gfx1250) — compile-verified
//
#include <hip/hip_runtime.h>
#include <cstdint>

// ComplexLayerNorm1d for B=8, C=256, T=8192 (fp32).
// Bandwidth-bound (~256 MB min traffic @ 23.3 TB/s ≈ 11 us). Strategy:
// stage each (256 x 64) x 2-plane tile HBM->LDS exactly once via the CDNA5
// Tensor Data Mover, run the channel-reduction pass and the whiten+affine
// pass out of LDS, and write outputs with b128 coalesced stores.

#define B_      8
#define C_      256
#define T_      8192
#define TILE_T  64
#define NTHR    256
#define EPS_    1e-6f

typedef unsigned int uint32;
typedef __attribute__((ext_vector_type(4))) unsigned int u32x4;
typedef __attribute__((ext_vector_type(8))) unsigned int u32x8;

union F4 { float4 v; float f[4]; };

// Issue one 2D TDM load: tile (tile_d1 rows x tile_d0 elems, 4B elems) from
// global (row stride stride_d0 elems) into LDS at lds_byte_addr.
// D# layout per CDNA5 ISA Ch.8 (verified in round-1 asm): group0 =
// {count/type/lds/global addr}, group1 = {data_size, tensor dims, tile dims,
// dim0 stride}. Groups 2/3 are NULL for <=2D tensors.
__device__ __forceinline__ void tdm_load_tile_2d(uint32 lds_byte_addr,
                                                 const float* gsrc,
                                                 uint32 tile_d0, uint32 tile_d1,
                                                 uint32 tensor_d0, uint32 tensor_d1,
                                                 unsigned long long stride_d0)
{
    unsigned long long ga = (unsigned long long)(uintptr_t)gsrc;
    u32x4 g0;
    g0[0] = 1u;                                            // count=1, user desc
    g0[1] = lds_byte_addr;                                 // lds_addr [63:32]
    g0[2] = (uint32)(ga & 0xFFFFFFFFull);                  // global_addr lo
    g0[3] = (uint32)((ga >> 32) & 0x01FFFFFFull)           // global_addr [56:32]
          | (2u << 30);                                    // type = 2 ("image")
    u32x8 g1;
    g1[0] = (2u << 16);                                    // data_size = 4B; wg_mask=0
    g1[1] = (tensor_d0 & 0xFFFFu) << 16;                   // tensor_dim0 [15:0]
    g1[2] = (tensor_d0 >> 16) | ((tensor_d1 & 0xFFFFu) << 16); // td0 hi | td1 lo
    g1[3] = (tensor_d1 >> 16) | (tile_d0 << 16);           // td1 hi | tile_dim0
    g1[4] = tile_d1 & 0xFFFFu;                             // tile_dim1 (tile_dim2=0)
    g1[5] = (uint32)(stride_d0 & 0xFFFFFFFFull);           // dim0 stride lo
    g1[6] = (uint32)((stride_d0 >> 32) & 0xFFFFull);       // dim0 stride hi
    g1[7] = 0u;                                            // dim1 stride unused (2D)
    asm volatile("tensor_load_to_lds %0, %1, null, null"
                 :: "s"(g0), "s"(g1)
                 : "memory");
}

__global__ void __launch_bounds__(NTHR)
cln_whiten_kernel(const float* __restrict__ xr, const float* __restrict__ xi,
                  const float* __restrict__ wt, const float* __restrict__ bs,
                  float* __restrict__ yr, float* __restrict__ yi)
{
    __shared__ __align__(16) float sm_xr[C_ * TILE_T];   // 64 KB, x_real tile
    __shared__ __align__(16) float sm_xi[C_ * TILE_T];   // 64 KB, x_imag tile
    __shared__ float sm_w[4 * C_];                       // weight (2,2,C)
    __shared__ float sm_b[2 * C_];                       // bias (2,C)
    __shared__ float sm_red[5 * NTHR];                   // partial sums
    __shared__ __align__(16) float sm_wst[5 * TILE_T];   // mu_r,mu_i,w00,w01,w11

    const int tid  = threadIdx.x;
    const int bid  = blockIdx.x;
    const int b    = bid >> 7;               // T_/TILE_T = 128 tiles per batch
    const int tile = bid & 127;
    const int t0   = tile * TILE_T;
    const size_t base = (size_t)b * (size_t)C_ * (size_t)T_ + (size_t)t0;

    // ---- Phase 1: TDM-stage the x tiles, prefetch weight/bias to LDS ----
    if (tid < 32) {  // first wave32 only: TDM issues per wave, EXEC-independent
        uint32 lds_r = (uint32)(unsigned long long)(uintptr_t)&sm_xr[0];
        uint32 lds_i = (uint32)(unsigned long long)(uintptr_t)&sm_xi[0];
        tdm_load_tile_2d(lds_r, xr + base, TILE_T, C_, (uint32)T_, (uint32)C_,
                         (unsigned long long)T_);
        tdm_load_tile_2d(lds_i, xi + base, TILE_T, C_, (uint32)T_, (uint32)C_,
                         (unsigned long long)T_);
    }
    for (int idx = tid; idx < 4 * C_; idx += NTHR) sm_w[idx] = wt[idx];
    for (int idx = tid; idx < 2 * C_; idx += NTHR) sm_b[idx] = bs[idx];
    if (tid < 32) __builtin_amdgcn_s_wait_tensorcnt(0);
    __syncthreads();

    // ---- Phase 2: channel reduction (4 threads per t-column) ----
    const int tl = tid & (TILE_T - 1);       // t within tile, 0..63
    const int q  = tid >> 6;                 // channel quarter, 0..3

    float sr = 0.f, si = 0.f, srr = 0.f, sii = 0.f, sri = 0.f;
    const int c0 = q * (C_ / 4);
    #pragma unroll 8
    for (int cc = 0; cc < C_ / 4; ++cc) {
        float r = sm_xr[(c0 + cc) * TILE_T + tl];
        float m = sm_xi[(c0 + cc) * TILE_T + tl];
        sr += r;  si += m;
        srr = fmaf(r, r, srr);
        sii = fmaf(m, m, sii);
        sri = fmaf(r, m, sri);
    }
    sm_red[0 * NTHR + tid] = sr;
    sm_red[1 * NTHR + tid] = si;
    sm_red[2 * NTHR + tid] = srr;
    sm_red[3 * NTHR + tid] = sii;
    sm_red[4 * NTHR + tid] = sri;
    __syncthreads();

    if (q == 0) {                            // tids 0..63: one per t-column
        float Sr = 0.f, Si = 0.f, Srr = 0.f, Sii = 0.f, Sri = 0.f;
        #pragma unroll
        for (int p = 0; p < 4; ++p) {
            int o = p * TILE_T + tl;
            Sr  += sm_red[0 * NTHR + o];
            Si  += sm_red[1 * NTHR + o];
            Srr += sm_red[2 * NTHR + o];
            Sii += sm_red[3 * NTHR + o];
            Sri += sm_red[4 * NTHR + o];
        }
        const float invC = 1.0f / (float)C_;
        float mr = Sr * invC, mi = Si * invC;
        float a  = fmaf(-mr, mr, Srr * invC) + EPS_;   // Vrr + eps
        float d  = fmaf(-mi, mi, Sii * invC) + EPS_;   // Vii + eps
        float bb = fmaf(-mr, mi, Sri * invC);          // Vri
        // symmetric SPD inverse sqrt, closed form (== eigh path):
        // s=sqrt(det), t=sqrt(tr+2s), V^-1/2 = [[d+s,-b],[-b,a+s]]/(s*t)
        float det = fmaxf(fmaf(a, d, -bb * bb), 0.0f);
        float s   = sqrtf(det);
        float tt  = sqrtf(a + d + 2.0f * s);
        float inv = 1.0f / fmaxf(s * tt, 1e-20f);
        sm_wst[0 * TILE_T + tl] = mr;
        sm_wst[1 * TILE_T + tl] = mi;
        sm_wst[2 * TILE_T + tl] = (d + s) * inv;       // w00
        sm_wst[3 * TILE_T + tl] = -bb * inv;           // w01 == w10
        sm_wst[4 * TILE_T + tl] = (a + s) * inv;       // w11
    }
    __syncthreads();

    // ---- Phase 3: whiten + per-channel affine, b128 coalesced stores ----
    // Remap: each thread owns 4 consecutive t (tl4) and 16 channels (cq).
    const int tl4 = (tid & 15) * 4;          // t offset of a float4 group
    const int cq  = tid >> 4;                // channel group, 0..15

    F4 mr4, mi4, w004, w014, w114;
    mr4.v  = *(const float4*)&sm_wst[0 * TILE_T + tl4];
    mi4.v  = *(const float4*)&sm_wst[1 * TILE_T + tl4];
    w004.v = *(const float4*)&sm_wst[2 * TILE_T + tl4];
    w014.v = *(const float4*)&sm_wst[3 * TILE_T + tl4];
    w114.v = *(const float4*)&sm_wst[4 * TILE_T + tl4];

    #pragma unroll 2
    for (int cc = 0; cc < C_ / 16; ++cc) {
        const int c = cq * 16 + cc;
        F4 r4, i4, or4, oi4;
        r4.v = *(const float4*)&sm_xr[c * TILE_T + tl4];
        i4.v = *(const float4*)&sm_xi[c * TILE_T + tl4];
        // y_j = sum_i z_i * weight[i][j][c] + bias[j][c]
        const float a00 = sm_w[0 * C_ + c];  // W[0,0,c]
        const float a01 = sm_w[1 * C_ + c];  // W[0,1,c]
        const float a10 = sm_w[2 * C_ + c];  // W[1,0,c]
        const float a11 = sm_w[3 * C_ + c];  // W[1,1,c]
        const float b0  = sm_b[c];
        const float b1  = sm_b[C_ + c];
        #pragma unroll
        for (int j = 0; j < 4; ++j) {
            float r  = r4.f[j] - mr4.f[j];
            float m  = i4.f[j] - mi4.f[j];
            float zr = fmaf(r, w004.f[j], m * w014.f[j]);
            float zi = fmaf(r, w014.f[j], m * w114.f[j]);
            or4.f[j] = fmaf(zr, a00, fmaf(zi, a10, b0));
            oi4.f[j] = fmaf(zr, a01, fmaf(zi, a11, b1));
        }
        size_t o = base + (size_t)c * T_ + (size_t)tl4;
        *(float4*)&yr[o] = or4.v;
        *(float4*)&yi[o] = oi4.v;
    }
}

extern "C" void kernel_launch(void* const* d_in, const int* in_sizes, int n_in,
                              void* d_out, int out_size, void* d_ws, size_t ws_size,
                              hipStream_t stream) {
    (void)in_sizes; (void)n_in; (void)out_size; (void)d_ws; (void)ws_size;
    const float* xr = (const float*)d_in[0];   // (B,C,T) f32
    const float* xi = (const float*)d_in[1];   // (B,C,T) f32
    const float* wt = (const float*)d_in[2];   // (2,2,C) f32
    const float* bs = (const float*)d_in[3];   // (2,C)   f32
    float* yr = (float*)d_out;                              // output 0 flat
    float* yi = yr + (size_t)B_ * (size_t)C_ * (size_t)T_;  // output 1 flat

    const int nblocks = B_ * (T_ / TILE_T);   // 8 * 128 = 1024
    cln_whiten_kernel<<<dim3(nblocks), dim3(NTHR), 0, stream>>>(
        xr, xi, wt, bs, yr, yi);
}